// QModel_65481071408560
// MI455X (gfx1250) — compile-verified
//
#include <hip/hip_runtime.h>

// ---------------------------------------------------------------------------
// 12-qubit batched statevector simulator for MI455X (gfx1250).
//
// One 256-thread workgroup per batch element; full 4096-amplitude complex
// state resident in LDS as a single array sSt[2*DIM] (re at 0, im at DIM) so
// all re/im selection is branchless index arithmetic. Layer 1q gates are
// fused 3-wires-at-a-time into 16x16 real-ified matrices and applied with
// V_WMMA_F32_16X16X4_F32 (K=16 via 4 accumulating WMMAs; 32 column tiles of
// 16 amplitude-groups). CNOT chains collapse to one suffix-XOR permutation
// per layer. Expvals + linear head fold into one weighted |amp|^2 reduction.
// ---------------------------------------------------------------------------

typedef float v2f __attribute__((ext_vector_type(2)));
typedef float v8f __attribute__((ext_vector_type(8)));

#define NW   12
#define DIM  4096          // 2^12
#define NT   256           // threads per block (8 wave32s)
#define APT  (DIM / NT)    // amplitudes per thread = 16

// (a+bi)*(c+di)*(e+fi) -> (pr, pi)
__device__ __forceinline__ void cmul3(float ar, float ai, float br, float bi,
                                      float cr, float ci, float& pr, float& pi) {
  float xr = ar * br - ai * bi;
  float xi = ar * bi + ai * br;
  pr = xr * cr - xi * ci;
  pi = xr * ci + xi * cr;
}

__global__ __launch_bounds__(NT) void qsim_kernel(
    const float* __restrict__ feats,    // (B, 12)
    const float* __restrict__ params,   // (3, 12)
    const float* __restrict__ headw,    // (12)
    const float* __restrict__ headb,    // (1)
    float* __restrict__ out)            // (B)
{
  __shared__ float sSt[2 * DIM];        // 32 KB: [0,DIM)=re, [DIM,2*DIM)=im
  __shared__ float sMat[12][256];       // 12 fused 16x16 real-ified gates, 12 KB
  __shared__ float sWG[3][12][8];       // per-(layer,wire) complex 2x2 = RY*RX
  __shared__ float sFc[NW], sFs[NW], sHw[NW];
  __shared__ float sRed[NT];

  const int tid = threadIdx.x;
  const int b   = blockIdx.x;

  // ---- per-wire layer gates U = RY(p)*RX(p), shared across batch -----------
  // RY=[[c,-s],[s,c]], RX=[[c,-is],[-is,c]] =>
  // U = [[c^2 + i s^2, -cs(1+i)], [cs(1-i), c^2 - i s^2]]
  if (tid < 36) {
    int s = tid / 12, w = tid % 12;
    float p  = params[s * 12 + w];
    float c  = __cosf(0.5f * p);
    float sn = __sinf(0.5f * p);
    float c2 = c * c, s2 = sn * sn, cs = c * sn;
    float* g = sWG[s][w];
    g[0] = c2;  g[1] = s2;     // U00
    g[2] = -cs; g[3] = -cs;    // U01
    g[4] = cs;  g[5] = -cs;    // U10
    g[6] = c2;  g[7] = -s2;    // U11
  }
  if (tid < NW) {
    float f = feats[b * NW + tid];
    sFc[tid] = __cosf(0.5f * f);
    sFs[tid] = __sinf(0.5f * f);
    sHw[tid] = headw[tid];
  }
  __syncthreads();

  // ---- build 12 fused gates: kron(U_{3g}, U_{3g+1}, U_{3g+2}) real-ified ---
  // 12 matrices x 64 complex entries = 768 entries.
  for (int e = tid; e < 768; e += NT) {
    int mat = e >> 6;            // 0..11  (= layer*4 + group)
    int s   = mat >> 2;
    int g   = mat & 3;
    int ent = e & 63;
    int r = ent >> 3, c = ent & 7;
    const float* Ua = sWG[s][3 * g + 0];
    const float* Ub = sWG[s][3 * g + 1];
    const float* Uc = sWG[s][3 * g + 2];
    int ia = (((r >> 2) & 1) * 2 + ((c >> 2) & 1)) * 2;
    int ib = (((r >> 1) & 1) * 2 + ((c >> 1) & 1)) * 2;
    int ic = (((r     ) & 1) * 2 + ((c     ) & 1)) * 2;
    float pr, pi;
    cmul3(Ua[ia], Ua[ia + 1], Ub[ib], Ub[ib + 1], Uc[ic], Uc[ic + 1], pr, pi);
    float* M = sMat[mat];
    M[r * 16 + c]            = pr;    // Gr
    M[r * 16 + c + 8]        = -pi;   // -Gi
    M[(r + 8) * 16 + c]      = pi;    // Gi
    M[(r + 8) * 16 + c + 8]  = pr;    // Gr
  }

  // ---- RY-encoded product state (skips the 12 encoding gates) --------------
  // wire w <-> bit (11-w); RY(f)|0> = [cos f/2, sin f/2]
  for (int i = tid; i < DIM; i += NT) {
    float prod = 1.0f;
    #pragma unroll
    for (int w = 0; w < NW; ++w)
      prod *= ((i >> (11 - w)) & 1) ? sFs[w] : sFc[w];
    sSt[i]       = prod;
    sSt[DIM + i] = 0.0f;
  }
  __syncthreads();

  const int lane = tid & 31;
  const int half = lane >> 4;     // 0: lanes 0-15, 1: lanes 16-31
  const int nrow = lane & 15;     // A row / B,D column within tile
  // wave id is uniform within a wave32: pin it to an SGPR so the tile loop is
  // scalar (no per-iteration EXEC masking; WMMA's EXEC==all-1s is structural).
  const int wave = __builtin_amdgcn_readfirstlane(tid >> 5);   // 8 waves

  for (int layer = 0; layer < 3; ++layer) {
    // ---- 4 fused 3-wire gate stages, each Y = M(16x16) * X(16x512) --------
    // Unroll the stage loop so plo/lowmask are compile-time constants (gives
    // constant-offset paired ds_load/store_2addr codegen); keep the tile loop
    // rolled — it is scalar thanks to the SGPR wave id.
    #pragma unroll
    for (int g = 0; g < 4; ++g) {
      const float* M = sMat[layer * 4 + g];
      const int plo = 9 - 3 * g;               // low bit of the 3-bit group
      const int lowmask = (1 << plo) - 1;
      for (int tile = wave; tile < 32; tile += 8) {
        int col  = tile * 16 + nrow;           // group (column) 0..511
        int base = ((col >> plo) << (plo + 3)) | (col & lowmask);
        v8f d = {0.f, 0.f, 0.f, 0.f, 0.f, 0.f, 0.f, 0.f};
        #pragma unroll
        for (int kc = 0; kc < 4; ++kc) {
          int k0 = kc * 4 + half * 2;          // A/B K layout: V0->{K0,K2},V1->{K1,K3}
          v2f a, x;
          a.x = M[nrow * 16 + k0];
          a.y = M[nrow * 16 + k0 + 1];
          // X rows 0-7 = re of sub-amp k, rows 8-15 = im (uniform per kc
          // after unroll: kc<2 -> re half, kc>=2 -> im half).
          int srcoff = ((kc < 2) ? 0 : DIM) + base;
          int ks = k0 & 7;
          x.x = sSt[srcoff + (ks << plo)];
          x.y = sSt[srcoff + ((ks + 1) << plo)];
          d = __builtin_amdgcn_wmma_f32_16x16x4_f32(
              false, a, false, x, (short)0, d, false, false);
        }
        // D: VGPR r holds row (r + 8*half); rows 0-7 -> re, 8-15 -> im.
        // (r + 8*half) & 7 == r, so only the re/im base differs per half:
        // branchless via half*DIM index offset.
        int dstoff = half * DIM + base;
        #pragma unroll
        for (int r = 0; r < 8; ++r)
          sSt[dstoff + (r << plo)] = d[r];
      }
      __syncthreads();
    }

    // ---- CNOT chain w=0..10 == suffix-XOR permutation of the 12-bit index -
    // new_bit_j = XOR(bits j..11); in-place via registers + barrier.
    float rr[APT], ri[APT];
    #pragma unroll
    for (int t = 0; t < APT; ++t) {
      int i = tid + NT * t;
      rr[t] = sSt[i]; ri[t] = sSt[DIM + i];
    }
    __syncthreads();
    #pragma unroll
    for (int t = 0; t < APT; ++t) {
      int i = tid + NT * t;
      int j = i;
      j ^= j >> 1; j ^= j >> 2; j ^= j >> 4; j ^= j >> 8;
      j &= 0xFFF;
      sSt[j] = rr[t]; sSt[DIM + j] = ri[t];
    }
    __syncthreads();
  }

  // ---- expvals + head folded: out = b + sum_i |amp|^2 * sum_w hw[w]*sgn ----
  float acc = 0.0f;
  for (int t = 0; t < APT; ++t) {
    int i = tid + NT * t;
    float pr = sSt[i], pi = sSt[DIM + i];
    float p = pr * pr + pi * pi;
    float coef = 0.0f;
    #pragma unroll
    for (int w = 0; w < NW; ++w) {
      float hw = sHw[w];
      coef += ((i >> (11 - w)) & 1) ? -hw : hw;
    }
    acc += p * coef;
  }
  sRed[tid] = acc;
  __syncthreads();
  for (int off = NT / 2; off > 0; off >>= 1) {
    if (tid < off) sRed[tid] += sRed[tid + off];
    __syncthreads();
  }
  if (tid == 0) out[b] = sRed[0] + headb[0];
}

extern "C" void kernel_launch(void* const* d_in, const int* in_sizes, int n_in,
                              void* d_out, int out_size, void* d_ws, size_t ws_size,
                              hipStream_t stream) {
  (void)n_in; (void)d_ws; (void)ws_size;
  const float* feats  = (const float*)d_in[0];  // state_batch (B,12)
  const float* params = (const float*)d_in[1];  // (3,12)
  const float* headw  = (const float*)d_in[2];  // (1,12)
  const float* headb  = (const float*)d_in[3];  // (1,)
  float* out = (float*)d_out;
  int B = in_sizes[0] / NW;                     // 512
  (void)out_size;
  qsim_kernel<<<B, NT, 0, stream>>>(feats, params, headw, headb, out);
}